// AttnCutLoss_52312701665762
// MI455X (gfx1250) — compile-verified
//
#include <hip/hip_runtime.h>
#include <hip/hip_bf16.h>
#include <stdint.h>

#define S_LEN   4096
#define TPB     256
#define EPT     16                      // elements per thread (TPB*EPT == S_LEN)
#define INV_TAU 1.0526315789473684f     // 1/0.95

typedef __attribute__((ext_vector_type(2))) float v2f;
typedef __attribute__((ext_vector_type(8))) float v8f;

// One block (256 threads = 8 wave32) per row.
// partial[row] = (sum_j log(out[row,j])) / (sum_j exp(r_j/tau)),  r_j = 2*tp_j/(T+j+1)
__global__ __launch_bounds__(TPB) void attncut_row_kernel(
    const float* __restrict__ out, const float* __restrict__ lab,
    float* __restrict__ partial)
{
    __shared__ float s_lab[S_LEN];   // 16 KB staged label row
    __shared__ float s_wave[8];
    __shared__ float s_total;
    __shared__ float s_q[8];
    __shared__ float s_l[8];

    const int tid  = threadIdx.x;
    const int lane = tid & 31;
    const int wid  = tid >> 5;
    const int row  = blockIdx.x;

    // ---- async DMA of the full label row into LDS (ASYNCcnt path) ----
    const float* grow = lab + (size_t)row * S_LEN;
    unsigned lds_base = (unsigned)(uintptr_t)&s_lab[0];
#pragma unroll
    for (int it = 0; it < 4; ++it) {
        unsigned off   = (unsigned)(tid * 16 + it * 4096);   // byte offset in row
        unsigned laddr = lds_base + off;                     // mirrored LDS address
        asm volatile("global_load_async_to_lds_b128 %0, %1, %2"
                     :: "v"(laddr), "v"(off), "s"(grow) : "memory");
    }
    asm volatile("s_wait_asynccnt 0" ::: "memory");
    __syncthreads();

    // ---- pull this thread's 16 labels into registers (ds_load_b128 x4) ----
    float l[EPT];
    {
        const float4* lv = (const float4*)(s_lab + tid * EPT);
#pragma unroll
        for (int v = 0; v < 4; ++v) {
            float4 t = lv[v];
            l[4*v+0] = t.x; l[4*v+1] = t.y; l[4*v+2] = t.z; l[4*v+3] = t.w;
        }
    }
    float local = 0.f;
#pragma unroll
    for (int j = 0; j < EPT; ++j) local += l[j];

    // ---- wave32 inclusive scan of per-thread sums ----
    float incl = local;
#pragma unroll
    for (int o = 1; o < 32; o <<= 1) {
        float n = __shfl_up(incl, o, 32);
        if (lane >= o) incl += n;
    }
    if (lane == 31) s_wave[wid] = incl;
    __syncthreads();
    if (tid == 0) {                       // scan the 8 wave totals
        float acc = 0.f;
#pragma unroll
        for (int w = 0; w < 8; ++w) { float t = s_wave[w]; s_wave[w] = acc; acc += t; }
        s_total = acc;
    }
    __syncthreads();

    const float T  = s_total;
    float tp       = s_wave[wid] + (incl - local);   // exclusive prefix for this thread

    // ---- stream this thread's 16 outputs (global_load_b128 x4) ----
    float ov[EPT];
    {
        const float4* og = (const float4*)(out + (size_t)row * S_LEN + (size_t)tid * EPT);
#pragma unroll
        for (int v = 0; v < 4; ++v) {
            float4 t = og[v];
            ov[4*v+0] = t.x; ov[4*v+1] = t.y; ov[4*v+2] = t.z; ov[4*v+3] = t.w;
        }
    }

    // ---- fused F1 + softmax-weight + log accumulation ----
    float qs = 0.f, ls = 0.f;
    const float kb = T + (float)(tid * EPT);         // T + base index
#pragma unroll
    for (int j = 0; j < EPT; ++j) {
        tp += l[j];                                   // inclusive tp at cutoff
        float r = (2.0f * tp) / (kb + (float)(j + 1));// F1@k == 2*tp/(T+k), exact
        qs += __expf(r * INV_TAU);                    // v_exp_f32
        ls += __logf(ov[j]);                          // v_log_f32
    }

    // ---- deterministic block reduction of (qs, ls) ----
#pragma unroll
    for (int o = 16; o > 0; o >>= 1) {
        qs += __shfl_down(qs, o, 32);
        ls += __shfl_down(ls, o, 32);
    }
    if (lane == 0) { s_q[wid] = qs; s_l[wid] = ls; }
    __syncthreads();
    if (tid == 0) {
        float q = 0.f, s = 0.f;
#pragma unroll
        for (int w = 0; w < 8; ++w) { q += s_q[w]; s += s_l[w]; }
        partial[row] = s / q;                         // row's  sum_log / norm
    }
}

// Reduce B row-partials to the scalar loss. The 256 thread-local sums are
// reduced on wave 0 with v_wmma_f32_16x16x4_f32 against a ones A-matrix:
// each WMMA accumulates 64 values (D[0,n] = column sums), 4 WMMAs cover 256.
__global__ __launch_bounds__(TPB) void attncut_final_kernel(
    const float* __restrict__ partial, float* __restrict__ loss,
    int n, float scale)
{
    __shared__ float s_red[TPB];
    const int tid = threadIdx.x;

    float acc = 0.f;
    for (int i = tid; i < n; i += TPB) acc += partial[i];  // fixed order
    s_red[tid] = acc;
    __syncthreads();

    if (tid < 32) {                      // wave 0 only: EXEC all ones for WMMA
        v2f a; a.x = 1.0f; a.y = 1.0f;   // A = ones 16x4
        v8f c = {};
#pragma unroll
        for (int it = 0; it < 4; ++it) { // pack 64 values into B = 4x16
            v2f b;
            b.x = s_red[it * 64 + tid];
            b.y = s_red[it * 64 + 32 + tid];
            c = __builtin_amdgcn_wmma_f32_16x16x4_f32(
                    false, a, false, b, (short)0, c, false, false);
        }
        float t = c[0];                  // row 0 of D = column sums
#pragma unroll
        for (int o = 8; o > 0; o >>= 1)  // sum the 16 column sums
            t += __shfl_xor(t, o, 32);
        if (tid == 0) loss[0] = t * scale;
    }
}

extern "C" void kernel_launch(void* const* d_in, const int* in_sizes, int n_in,
                              void* d_out, int out_size, void* d_ws, size_t ws_size,
                              hipStream_t stream) {
    (void)n_in; (void)out_size; (void)ws_size;
    const float* out = (const float*)d_in[0];   // [B,S,1] f32
    const float* lab = (const float*)d_in[1];   // [B,S]   f32
    float* loss    = (float*)d_out;             // scalar
    float* partial = (float*)d_ws;              // B floats of scratch

    const int B = in_sizes[1] / S_LEN;

    attncut_row_kernel<<<B, TPB, 0, stream>>>(out, lab, partial);
    attncut_final_kernel<<<1, TPB, 0, stream>>>(partial, loss, B, -1.0f / (float)B);
}